// DetectionLoss_81862076662063
// MI455X (gfx1250) — compile-verified
//
#include <hip/hip_runtime.h>
#include <hip/hip_bf16.h>
#include <math.h>
#include <stdint.h>

#define BATCH 32
#define NBOX  20
#define CNUM  3
#define PIX   128            // pixels per phaseA block
#define TPB   (PIX * 3)      // 384 threads = 12 waves

typedef __attribute__((ext_vector_type(2))) float    v2f;
typedef __attribute__((ext_vector_type(8))) float    v8f;
typedef __attribute__((ext_vector_type(4))) unsigned u32x4;
typedef __attribute__((ext_vector_type(8))) int      i32x8;
typedef __attribute__((ext_vector_type(4))) int      i32x4;

// ---------------------------------------------------------------------------
// Wave32 sum on the CDNA5 matrix pipe (V_WMMA_F32_16X16X4_F32, B = ones).
// Requires EXEC == all ones at the call site (uniform control flow).
// ---------------------------------------------------------------------------
__device__ __forceinline__ float wave_sum32(float x) {
  v2f a; a[0] = x;    a[1] = 0.0f;
  v2f b; b[0] = 1.0f; b[1] = 1.0f;
  v8f c = {};
  c = __builtin_amdgcn_wmma_f32_16x16x4_f32(false, a, false, b, (short)0, c,
                                            false, false);
  float s = ((c[0] + c[1]) + (c[2] + c[3])) + ((c[4] + c[5]) + (c[6] + c[7]));
  s += __shfl_xor(s, 16, 32);
  return s;
}

__device__ __forceinline__ float block_sum(float v, float* sred, int tid, int nwarp) {
  float w = wave_sum32(v);
  int warp = tid >> 5, lane = tid & 31;
  __syncthreads();
  if (lane == 0) sred[warp] = w;
  __syncthreads();
  float total = 0.0f;
  for (int i = 0; i < nwarp; ++i) total += sred[i];
  return total;
}

__device__ __forceinline__ float bce_logits(float x, float t) {
  return fmaxf(x, 0.0f) - x * t + log1pf(expf(-fabsf(x)));
}

// ---------------------------------------------------------------------------
// Phase A: one block = 128 pixels x 3 anchors of one (scale, batch).
// Pred tile (24 ch x 128 px, row stride H*W) is DMA'd into LDS by the TDM.
// Boxes/labels staged via the async-to-LDS copy engine (ASYNCcnt path).
// accum[(s*32+b)*8+q]: q0=pos_sum q1=all_sum q2=cls q3=loc q4=npos q5=nneg q6=obj
// ---------------------------------------------------------------------------
__global__ __launch_bounds__(TPB) void phaseA(
    const float* __restrict__ pred, const float* __restrict__ anchors,
    const float* __restrict__ boxes, const int* __restrict__ labels,
    float* __restrict__ accum, float* __restrict__ negvals,
    int H, int W, int sIdx)
{
  __shared__ float spred[24 * PIX];   // 12 KB tile: [channel][pixel]
  __shared__ float sbox[NBOX * 4];
  __shared__ float sarea[NBOX];
  __shared__ int   slab[NBOX];
  __shared__ float sred[TPB / 32];

  const int tid = threadIdx.x;
  const int b   = blockIdx.y;
  const int HW  = H * W;
  const int A   = HW * 3;
  const int p0  = blockIdx.x * PIX;
  const int a   = p0 * 3 + tid;            // global anchor id (grid exact)

  // ---- stage boxes + labels via async copy engine (tracked by ASYNCcnt) ----
  {
    const unsigned voff = (unsigned)tid * 4u;
    if (tid < NBOX * 4) {
      unsigned lbox = (unsigned)(uintptr_t)(&sbox[0]) + voff;
      unsigned long long gb =
          (unsigned long long)(uintptr_t)(boxes + (size_t)b * NBOX * 4);
      asm volatile("global_load_async_to_lds_b32 %0, %1, %2"
                   :: "v"(lbox), "v"(voff), "s"(gb) : "memory");
    }
    if (tid < NBOX) {
      unsigned llab = (unsigned)(uintptr_t)(&slab[0]) + voff;
      unsigned long long gl =
          (unsigned long long)(uintptr_t)(labels + (size_t)b * NBOX);
      asm volatile("global_load_async_to_lds_b32 %0, %1, %2"
                   :: "v"(llab), "v"(voff), "s"(gl) : "memory");
    }
    asm volatile("s_wait_asynccnt 0x0" ::: "memory");
  }

  // ---- stage the 24 x PIX pred tile via the Tensor Data Mover ----
  if (tid < 32) {                                 // wave 0 issues the DMA
    const unsigned long long ga =
        (unsigned long long)(uintptr_t)(pred + ((size_t)b * 24) * HW + p0);
    u32x4 g0;
    g0[0] = 1u;                                   // count=1, user descriptor
    g0[1] = (unsigned)(uintptr_t)(&spred[0]);     // lds_addr
    g0[2] = (unsigned)ga;                         // global_addr[31:0]
    g0[3] = (unsigned)(ga >> 32) | 0x80000000u;   // global_addr[56:32] | type=2
    i32x8 g1;
    g1[0] = (int)(2u << 16);                      // data_size = 4 bytes
    g1[1] = (int)(((unsigned)HW & 0xFFFFu) << 16);// tensor_dim0[15:0]
    g1[2] = (int)(((unsigned)HW >> 16) | (24u << 16)); // dim0[31:16] | dim1[15:0]
    g1[3] = (int)((unsigned)PIX << 16);           // dim1[31:16]=0 | tile_dim0
    g1[4] = 24;                                   // tile_dim1 | tile_dim2=0
    g1[5] = HW;                                   // tensor_dim0_stride[31:0]
    g1[6] = 0;                                    // stride0 hi | stride1 lo
    g1[7] = 0;
    i32x4 gz = {0, 0, 0, 0};
#if __clang_major__ >= 23
    i32x8 gz8 = {0, 0, 0, 0, 0, 0, 0, 0};
    __builtin_amdgcn_tensor_load_to_lds(g0, g1, gz, gz, gz8, 0);
#else
    __builtin_amdgcn_tensor_load_to_lds(g0, g1, gz, gz, 0);
#endif
    __builtin_amdgcn_s_wait_tensorcnt(0);
  }
  __builtin_prefetch(pred + ((size_t)b * 24) * HW + p0 + PIX, 0, 0);
  __syncthreads();

  if (tid < NBOX)
    sarea[tid] = (sbox[tid*4+2] - sbox[tid*4+0]) * (sbox[tid*4+3] - sbox[tid*4+1]);
  __syncthreads();

  // ---- IoU max / argmax over the 20 boxes ----
  const float ax1 = anchors[a*4+0], ay1 = anchors[a*4+1];
  const float ax2 = anchors[a*4+2], ay2 = anchors[a*4+3];
  const float aa  = (ax2 - ax1) * (ay2 - ay1);

  float best_iou = -1.0f; int best = 0;
#pragma unroll
  for (int n = 0; n < NBOX; ++n) {
    float bx1 = sbox[n*4+0], by1 = sbox[n*4+1], bx2 = sbox[n*4+2], by2 = sbox[n*4+3];
    float iw = fmaxf(fminf(ax2, bx2) - fmaxf(ax1, bx1), 0.0f);
    float ih = fmaxf(fminf(ay2, by2) - fmaxf(ay1, by1), 0.0f);
    float inter = iw * ih;
    float iou = inter / (aa + sarea[n] - inter + 1e-8f);
    if (iou > best_iou) { best_iou = iou; best = n; }   // first-max == jnp.argmax
  }
  const bool pos = best_iou >= 0.5f;
  const bool neg = best_iou <  0.3f;

  // ---- per-anchor terms, pred read from the LDS tile ----
  const int lp = tid / 3, sa = tid - lp * 3;      // local pixel, anchor-size idx
  const float pb0 = spred[(sa*8+0)*PIX + lp];
  const float pb1 = spred[(sa*8+1)*PIX + lp];
  const float pb2 = spred[(sa*8+2)*PIX + lp];
  const float pb3 = spred[(sa*8+3)*PIX + lp];
  const float po  = spred[(sa*8+4)*PIX + lp];
  const float z0  = spred[(sa*8+5)*PIX + lp];
  const float z1  = spred[(sa*8+6)*PIX + lp];
  const float z2  = spred[(sa*8+7)*PIX + lp];

  const float obj = bce_logits(po, pos ? 1.0f : 0.0f);

  float zm  = fmaxf(z0, fmaxf(z1, z2));
  float lse = zm + logf(expf(z0 - zm) + expf(z1 - zm) + expf(z2 - zm));
  int lab = slab[best] - 1;
  lab = lab < 0 ? 0 : (lab > CNUM - 1 ? CNUM - 1 : lab);
  float zt = (lab == 0) ? z0 : ((lab == 1) ? z1 : z2);
  float ce = lse - zt;

  float d0 = pb0 - sbox[best*4+0], d1 = pb1 - sbox[best*4+1];
  float d2 = pb2 - sbox[best*4+2], d3 = pb3 - sbox[best*4+3];
  float sl1 = 0.0f, ad;
  ad = fabsf(d0); sl1 += (ad < 1.0f) ? 0.5f * d0 * d0 : ad - 0.5f;
  ad = fabsf(d1); sl1 += (ad < 1.0f) ? 0.5f * d1 * d1 : ad - 0.5f;
  ad = fabsf(d2); sl1 += (ad < 1.0f) ? 0.5f * d2 * d2 : ad - 0.5f;
  ad = fabsf(d3); sl1 += (ad < 1.0f) ? 0.5f * d3 * d3 : ad - 0.5f;

  negvals[(size_t)b * A + a] = neg ? obj : -__builtin_inff();

  const float r0 = block_sum(pos ? obj  : 0.0f, sred, tid, TPB/32);
  const float r1 = block_sum(obj,               sred, tid, TPB/32);
  const float r2 = block_sum(pos ? ce   : 0.0f, sred, tid, TPB/32);
  const float r3 = block_sum(pos ? sl1  : 0.0f, sred, tid, TPB/32);
  const float r4 = block_sum(pos ? 1.0f : 0.0f, sred, tid, TPB/32);
  const float r5 = block_sum(neg ? 1.0f : 0.0f, sred, tid, TPB/32);
  if (tid == 0) {
    float* acc = accum + ((size_t)sIdx * BATCH + b) * 8;
    atomicAdd(acc + 0, r0); atomicAdd(acc + 1, r1); atomicAdd(acc + 2, r2);
    atomicAdd(acc + 3, r3); atomicAdd(acc + 4, r4); atomicAdd(acc + 5, r5);
  }
}

// ---------------------------------------------------------------------------
// Phase B: hard-negative top-k SUM via 31-step binary search over IEEE bit
// patterns (positive floats are order-isomorphic to their bit patterns).
// Rows stay L2-resident (8.3 MB total << 192 MB).
// ---------------------------------------------------------------------------
__global__ __launch_bounds__(1024) void phaseB(float* __restrict__ ws) {
  __shared__ float sred[32];
  const int tid = threadIdx.x;
  const int b = blockIdx.x;
  const int s = blockIdx.y;
  const int    Atab[3] = {49152, 12288, 3072};
  const size_t Btab[3] = {768,
                          768 + (size_t)BATCH * 49152,
                          768 + (size_t)BATCH * 49152 + (size_t)BATCH * 12288};
  const int A = Atab[s];
  float* acc = ws + ((size_t)s * BATCH + b) * 8;
  const float* __restrict__ vals = ws + Btab[s] + (size_t)b * A;

  const float pos_sum = acc[0];
  const float all_sum = acc[1];
  const int np = (int)(acc[4] + 0.5f);
  const int nn = (int)(acc[5] + 0.5f);
  int k = 3 * np; if (k > nn) k = nn;

  float objc;
  if (k <= 0) {                           // block-uniform branch
    objc = all_sum;
  } else {
    unsigned lo = 0u, hi = 0x7F800000u;   // [0, +inf) over positive floats
    for (int it = 0; it < 31; ++it) {
      const unsigned mid = (lo + hi) >> 1;
      const float t = __uint_as_float(mid);
      float cl = 0.0f;
      for (int i = tid; i < A; i += 1024)         // A divisible by 1024
        cl += (vals[i] >= t) ? 1.0f : 0.0f;
      const float cnt = block_sum(cl, sred, tid, 32);
      if (cnt >= (float)k) lo = mid; else hi = mid;
    }
    const float vk = __uint_as_float(lo);         // exact k-th largest value
    float sl = 0.0f, cg = 0.0f;
    for (int i = tid; i < A; i += 1024) {
      const float v = vals[i];
      if (v > vk) { sl += v; cg += 1.0f; }
    }
    const float sumGt = block_sum(sl, sred, tid, 32);
    const float cntGt = block_sum(cg, sred, tid, 32);
    objc = pos_sum + sumGt + ((float)k - cntGt) * vk;
  }
  if (tid == 0) acc[6] = objc;
}

// ---------------------------------------------------------------------------
// Phase C: final cross-(scale,batch) reduction + normalize.
// ---------------------------------------------------------------------------
__global__ __launch_bounds__(128) void phaseC(const float* __restrict__ ws,
                                              float* __restrict__ out) {
  __shared__ float sred[4];
  const int tid = threadIdx.x;
  float obj = 0.0f, cls = 0.0f, loc = 0.0f, npos = 0.0f;
  if (tid < 3 * BATCH) {
    const float* acc = ws + (size_t)tid * 8;
    obj = acc[6]; cls = acc[2]; loc = acc[3]; npos = acc[4];
  }
  const float tobj = block_sum(obj,  sred, tid, 4);
  const float tcls = block_sum(cls,  sred, tid, 4);
  const float tloc = block_sum(loc,  sred, tid, 4);
  const float tnp  = block_sum(npos, sred, tid, 4);
  if (tid == 0) {
    const float norm = fmaxf(1.0f, tnp);
    const float lo = tobj / norm, lc = tcls / norm, ll = tloc / norm;
    out[0] = lo; out[1] = lc; out[2] = ll; out[3] = lo + lc + 2.0f * ll;
  }
}

__global__ void initAcc(float* __restrict__ ws) {
  const int i = blockIdx.x * 256 + threadIdx.x;
  if (i < 768) ws[i] = 0.0f;
}

extern "C" void kernel_launch(void* const* d_in, const int* in_sizes, int n_in,
                              void* d_out, int out_size, void* d_ws, size_t ws_size,
                              hipStream_t stream) {
  (void)in_sizes; (void)n_in; (void)out_size; (void)ws_size;
  const float* pred0  = (const float*)d_in[0];
  const float* pred1  = (const float*)d_in[1];
  const float* pred2  = (const float*)d_in[2];
  const float* boxes  = (const float*)d_in[3];
  const int*   labels = (const int*)  d_in[4];
  const float* anc0   = (const float*)d_in[5];
  const float* anc1   = (const float*)d_in[6];
  const float* anc2   = (const float*)d_in[7];
  float* ws  = (float*)d_ws;
  float* out = (float*)d_out;

  const size_t NEG0 = 768;
  const size_t NEG1 = NEG0 + (size_t)BATCH * 49152;
  const size_t NEG2 = NEG1 + (size_t)BATCH * 12288;

  hipLaunchKernelGGL(initAcc, dim3(3), dim3(256), 0, stream, ws);
  hipLaunchKernelGGL(phaseA, dim3(16384 / PIX, BATCH), dim3(TPB), 0, stream,
                     pred0, anc0, boxes, labels, ws, ws + NEG0, 128, 128, 0);
  hipLaunchKernelGGL(phaseA, dim3(4096 / PIX, BATCH), dim3(TPB), 0, stream,
                     pred1, anc1, boxes, labels, ws, ws + NEG1, 64, 64, 1);
  hipLaunchKernelGGL(phaseA, dim3(1024 / PIX, BATCH), dim3(TPB), 0, stream,
                     pred2, anc2, boxes, labels, ws, ws + NEG2, 32, 32, 2);
  hipLaunchKernelGGL(phaseB, dim3(BATCH, 3), dim3(1024), 0, stream, ws);
  hipLaunchKernelGGL(phaseC, dim3(1), dim3(128), 0, stream, ws, out);
}